// Os2dHeadCreator_4337916969703
// MI455X (gfx1250) — compile-verified
//
#include <hip/hip_runtime.h>
#include <hip/hip_bf16.h>
#include <math.h>

// ---------------------------------------------------------------------------
// OS2D head for MI455X (gfx1250, wave32).
// Heavy op: corr GEMM  M=3600 (bC*15*15), K=1024, N=16384 (bA*64*64),
// 120.8 GFLOP f32 via V_WMMA_F32_16X16X4_F32 (native f32 matrix op — corr
// feeds log()/bilinear chains so reference f32 numerics are preserved).
// A and B are repacked as (k,k+1) pairs (B pre-scaled by 1/||f||) so the hot
// loop is pure b64 loads + WMMA: 9 global_load_b64 + 8 v_wmma per 4-k step.
// fm (64MB) + packed buffers are L2-resident (192MB L2) -> compute-bound.
// ---------------------------------------------------------------------------

typedef float v2f __attribute__((ext_vector_type(2)));
typedef float v8f __attribute__((ext_vector_type(8)));

#define BA 4
#define BC 16
#define DCH 1024
#define HA 64
#define WA 64
#define HT 15
#define WT 15
#define SPA (HA*WA)            // 4096
#define MROWS (BC*HT*WT)       // 3600
#define NCOLS (BA*SPA)         // 16384

// ---------------- 1/(||f||+eps) over channels, fm: [4][1024][64][64] -------
__global__ __launch_bounds__(256) void fm_norm_kernel(const float* __restrict__ fm,
                                                      float* __restrict__ inv_fm) {
    int n = blockIdx.x * 256 + threadIdx.x;      // n = a*4096 + s
    int a = n >> 12, s = n & (SPA - 1);
    const float* p = fm + (size_t)a * DCH * SPA + s;
    float acc = 0.f;
    for (int f = 0; f < DCH; ++f) { float v = p[(size_t)f * SPA]; acc += v * v; }
    inv_fm[n] = 1.f / (sqrtf(acc) + 1e-5f);
}

// ---------------- cf: [16][1024][15][15], pos p = b*225 + h*15 + w ---------
__global__ __launch_bounds__(256) void cf_norm_kernel(const float* __restrict__ cf,
                                                      float* __restrict__ inv_cf) {
    int p = blockIdx.x * 256 + threadIdx.x;
    if (p >= MROWS) return;
    int b = p / 225, r = p - b * 225;
    const float* q = cf + (size_t)b * DCH * 225 + r;
    float acc = 0.f;
    for (int f = 0; f < DCH; ++f) { float v = q[(size_t)f * 225]; acc += v * v; }
    inv_cf[p] = 1.f / (sqrtf(acc) + 1e-5f);
}

// -------- packed A: Atp[k2*7200 + m*2 + p], k = 2*k2+p, m = b*225+w*15+h ---
// (corr channel c = w*HT + h per the 'abwhxy' reshape in the reference)
__global__ __launch_bounds__(256) void build_Atp_kernel(const float* __restrict__ cf,
                                                        const float* __restrict__ inv_cf,
                                                        float* __restrict__ Atp) {
    int idx = blockIdx.x * 256 + threadIdx.x;    // 0..1024*3600-1
    int k = idx / MROWS;
    int m = idx - k * MROWS;
    int b = m / 225, c = m - b * 225;
    int wq = c / 15, hq = c - wq * 15;
    int pos = hq * 15 + wq;
    float v = cf[((size_t)(b * DCH + k)) * 225 + pos] * inv_cf[b * 225 + pos];
    Atp[(size_t)(k >> 1) * (2 * MROWS) + m * 2 + (k & 1)] = v;
}

// -------- packed+scaled B: Bp[((a*512 + k2)*4096 + s)*2 + p] ---------------
__global__ __launch_bounds__(256) void build_Bp_kernel(const float* __restrict__ fm,
                                                       const float* __restrict__ inv_fm,
                                                       float* __restrict__ Bp) {
    int idx = blockIdx.x * 256 + threadIdx.x;    // 0..4*1024*4096-1
    int s = idx & (SPA - 1);
    int k = (idx >> 12) & (DCH - 1);
    int a = idx >> 22;
    float v = fm[(size_t)idx] * inv_fm[a * SPA + s];   // fm is [a][k][s] contiguous
    Bp[(((size_t)(a * 512 + (k >> 1)) << 12) + s) * 2 + (k & 1)] = v;
}

// ---------------- WMMA f32 GEMM: corr[(a*3600+m)*4096 + s] -----------------
// One wave = 16(M) x 128(N): 8 v8f accumulators; 256 k-steps of 16x16x4.
// Fragments (ISA 7.12.2, 32-bit A 16x4): lanes0-15 K={k,k+1}, lanes16-31
// K={k+2,k+3}, M = m0+(lane&15); B mirrors with N = (lane&15).
// C/D: VGPR r -> M = r + 8*(lane>=16), N = lane&15.
__global__ __launch_bounds__(128) void corr_gemm_kernel(const float* __restrict__ Atp,
                                                        const float* __restrict__ Bp,
                                                        float* __restrict__ corr) {
    const int lane  = threadIdx.x & 31;
    const int wave  = threadIdx.x >> 5;
    const int m0    = blockIdx.x * 16;                // 225 M tiles
    const int n0    = blockIdx.y * 512 + wave * 128;  // 32 N groups of 512
    const int a     = n0 >> 12;                       // 128-wide tile never crosses a
    const int s0    = n0 & (SPA - 1);
    const int halfk = lane >> 4;                      // 0/1 -> K-pair select
    const int lr    = lane & 15;

    const float* Abase = Atp + (size_t)(m0 + lr) * 2;
    const float* Bbase = Bp + (((size_t)(a * 512) << 12) + s0 + lr) * 2;

    v8f acc[8] = {v8f{}, v8f{}, v8f{}, v8f{}, v8f{}, v8f{}, v8f{}, v8f{}};

#pragma unroll 2
    for (int k0 = 0; k0 < DCH; k0 += 4) {
        const int k2 = (k0 >> 1) + halfk;             // pair index for this half-wave
        v2f afrag = *(const v2f*)(Abase + (size_t)k2 * (2 * MROWS));
#pragma unroll
        for (int t = 0; t < 8; ++t) {
            v2f bfrag = *(const v2f*)(Bbase + (((size_t)k2 << 12) + t * 16) * 2);
            acc[t] = __builtin_amdgcn_wmma_f32_16x16x4_f32(
                false, afrag, false, bfrag, (short)0, acc[t], false, false);
        }
    }

    const int colbase = s0 + lr;
#pragma unroll
    for (int r = 0; r < 8; ++r) {
        int row = m0 + r + 8 * halfk;
        float* dst = corr + ((size_t)(a * MROWS + row)) * SPA + colbase;
#pragma unroll
        for (int t = 0; t < 8; ++t) dst[t * 16] = acc[t][r];
    }
}

// ---------------- head: theta, grids, loc, rec(=rec_det), corners ----------
__global__ __launch_bounds__(256) void head_kernel(const float* __restrict__ corr,
                                                   const float* __restrict__ w_align,
                                                   const float* __restrict__ b_align,
                                                   float* __restrict__ out) {
    __shared__ float sw[6 * 225];
    __shared__ float sb[6];
    for (int i = threadIdx.x; i < 6 * 225; i += 256) sw[i] = w_align[i];
    if (threadIdx.x < 6) sb[threadIdx.x] = b_align[threadIdx.x];
    __syncthreads();

    const int ab = blockIdx.y;                 // 0..63
    const int a = ab >> 4, b = ab & 15;
    const int s = blockIdx.x * 256 + threadIdx.x;   // 0..4095
    const int h = s >> 6, w = s & 63;

    const float* slab = corr + ((size_t)(a * MROWS + b * 225)) * SPA;
    const float* cbase = slab + s;

    float th[6] = {sb[0], sb[1], sb[2], sb[3], sb[4], sb[5]};
    for (int c = 0; c < 225; ++c) {
        float v = cbase[(size_t)c * SPA];      // lane-coalesced across the block
#pragma unroll
        for (int t = 0; t < 6; ++t) th[t] += v * sw[t * 225 + c];
    }

    float xmin = 1e30f, xmax = -1e30f, ymin = 1e30f, ymax = -1e30f;
    float pooled = 0.f;
    float corn[8];

    for (int i = 0; i < 15; ++i) {
        float gy = -1.f + (float)i * (2.f / 14.f);
#pragma unroll
        for (int j = 0; j < 15; ++j) {
            float gx = -1.f + (float)j * (2.f / 14.f);
            float lx = th[0] * gx + th[1] * gy + th[2];
            float ly = th[3] * gx + th[4] * gy + th[5];
            float xf = lx * 7.5f + (float)w;   // grids_fm ; grids_img = 16*grids_fm
            float yf = ly * 7.5f + (float)h;
            float xi = xf * 16.f, yi = yf * 16.f;
            xmin = fminf(xmin, xi); xmax = fmaxf(xmax, xi);
            ymin = fminf(ymin, yi); ymax = fmaxf(ymax, yi);
            if ((i == 0 || i == 14) && (j == 0 || j == 14)) {
                int k = ((i == 14) ? 4 : 0) + ((j == 14) ? 2 : 0);
                corn[k] = xi; corn[k + 1] = yi;
            }
            if (i >= 2 && i <= 12 && j >= 2 && j <= 12) {
                // gunit clip + resample collapses to clamp into [0,63]
                float x = fminf(fmaxf(xf, 0.f), 63.f);
                float y = fminf(fmaxf(yf, 0.f), 63.f);
                float x0 = floorf(x), y0 = floorf(y);
                float wx = x - x0,  wy = y - y0;
                int x0i = min(max((int)x0, 0), 63);
                int x1i = min(x0i + 1, 63);
                int y0i = min(max((int)y0, 0), 63);
                int y1i = min(y0i + 1, 63);
                const float* cc = slab + (size_t)(i + j * 15) * SPA;
                float v00 = cc[y0i * 64 + x0i], v01 = cc[y0i * 64 + x1i];
                float v10 = cc[y1i * 64 + x0i], v11 = cc[y1i * 64 + x1i];
                pooled += (v00 * (1.f - wx) + v01 * wx) * (1.f - wy)
                        + (v10 * (1.f - wx) + v11 * wx) * wy;
            }
        }
    }
    pooled *= (1.f / 121.f);

    // loc targets vs default box (center 16w,16h; size 240, >=1 so unclipped)
    float c_cx = 0.5f * (xmin + xmax);
    float c_cy = 0.5f * (ymin + ymax);
    float c_w  = fmaxf(xmax - xmin, 1.0f);     // clip_min_size keeps center
    float c_h  = fmaxf(ymax - ymin, 1.0f);
    const float inv240 = 1.f / 240.f;
    float loc0 = (c_cx - 16.f * (float)w) * inv240;
    float loc1 = (c_cy - 16.f * (float)h) * inv240;
    float loc2 = logf(c_w * inv240);
    float loc3 = logf(c_h * inv240);

    float* loc  = out;                    // (4,16,4,64,64)
    float* rec  = out + 1048576;          // (4,16,1,64,64)
    float* recd = out + 1310720;          // rec_det == rec in forward pass
    float* crn  = out + 1572864;          // (4,16,8,64,64)

    loc[((size_t)ab * 4 + 0) * SPA + s] = loc0;
    loc[((size_t)ab * 4 + 1) * SPA + s] = loc1;
    loc[((size_t)ab * 4 + 2) * SPA + s] = loc2;
    loc[((size_t)ab * 4 + 3) * SPA + s] = loc3;
    rec [(size_t)ab * SPA + s] = pooled;
    recd[(size_t)ab * SPA + s] = pooled;
#pragma unroll
    for (int k = 0; k < 8; ++k) crn[((size_t)ab * 8 + k) * SPA + s] = corn[k];
}

// ---------------------------------------------------------------------------
extern "C" void kernel_launch(void* const* d_in, const int* in_sizes, int n_in,
                              void* d_out, int out_size, void* d_ws, size_t ws_size,
                              hipStream_t stream) {
    const float* fm      = (const float*)d_in[0];  // (4,1024,64,64)
    const float* cf      = (const float*)d_in[1];  // (16,1024,15,15)
    const float* w_align = (const float*)d_in[2];  // (6,225)
    const float* b_align = (const float*)d_in[3];  // (6,)
    float* out = (float*)d_out;

    float* ws     = (float*)d_ws;
    float* inv_cf = ws;                        // 3600 floats
    float* inv_fm = ws + 4096;                 // 16384 floats
    float* Atp    = ws + 20480;                // 1024*3600  = 3,686,400 floats
    float* Bp     = ws + 3706880;              // 4*1024*4096 = 16,777,216 floats
    float* corr   = ws + 20484096;             // 4*3600*4096 = 58,982,400 floats

    fm_norm_kernel  <<<NCOLS / 256, 256, 0, stream>>>(fm, inv_fm);
    cf_norm_kernel  <<<(MROWS + 255) / 256, 256, 0, stream>>>(cf, inv_cf);
    build_Atp_kernel<<<(DCH * MROWS) / 256, 256, 0, stream>>>(cf, inv_cf, Atp);
    build_Bp_kernel <<<(BA * DCH * SPA) / 256, 256, 0, stream>>>(fm, inv_fm, Bp);
    corr_gemm_kernel<<<dim3(MROWS / 16, NCOLS / 512), 128, 0, stream>>>(Atp, Bp, corr);
    head_kernel     <<<dim3(SPA / 256, BA * BC), 256, 0, stream>>>(corr, w_align, b_align, out);
}